// HeteroGraphConv_33131377721484
// MI455X (gfx1250) — compile-verified
//
#include <hip/hip_runtime.h>
#include <cstdint>

typedef __attribute__((ext_vector_type(2))) float v2f;
typedef __attribute__((ext_vector_type(8))) float v8f;

#define HD 128

// ---------------------------------------------------------------------------
// Native CDNA5 no-return float atomic add (fire-and-forget, STOREcnt-tracked).
// Guarantees global_atomic_add_f32 regardless of the toolchain's atomicAdd
// lowering policy (avoids a CAS retry loop on the hot scatter path).
// ---------------------------------------------------------------------------
__device__ __forceinline__ void atom_add_f32(float* p, float v) {
  asm volatile("global_atomic_add_f32 %0, %1, off"
               :
               : "v"(p), "v"(v)
               : "memory");
}

// ---------------------------------------------------------------------------
// Zero-fill (float4 granularity; all our region sizes are multiples of 4)
// ---------------------------------------------------------------------------
__global__ void zero_f4_kernel(float4* __restrict__ p, long n4) {
  long i = (long)blockIdx.x * blockDim.x + threadIdx.x;
  long stride = (long)gridDim.x * blockDim.x;
  float4 z = make_float4(0.f, 0.f, 0.f, 0.f);
  for (; i < n4; i += stride) p[i] = z;
}

// ---------------------------------------------------------------------------
// Edge scatter: one wave (32 lanes) per edge; each lane handles 4 channels.
// accum[dst] += feat[src]; optionally cnt[dst] += 1 (for mean).
// ---------------------------------------------------------------------------
__global__ void scatter_add_kernel(const float* __restrict__ feat,
                                   const int* __restrict__ src,
                                   const int* __restrict__ dst,
                                   float* __restrict__ accum,
                                   float* __restrict__ cnt,
                                   int nE) {
  int lane = threadIdx.x & 31;
  int wid = (int)blockIdx.x * (blockDim.x >> 5) + (threadIdx.x >> 5);
  if (wid >= nE) return;
  int s = src[wid];
  int d = dst[wid];
  const float* fs = feat + (long)s * HD;
  float* fd = accum + (long)d * HD;
#pragma unroll
  for (int c = 0; c < HD; c += 32) {
    atom_add_f32(&fd[c + lane], fs[c + lane]);
  }
  if (cnt != nullptr && lane == 0) atom_add_f32(&cnt[d], 1.0f);
}

// ---------------------------------------------------------------------------
// Router update: out = feat + relu([h1|h2] @ W_r + b_r)
// io holds h1 on entry (accumulated in d_out), receives output in place.
// One wave per 16-row tile; 8 N-tiles of 16 cols held as accumulators;
// K streamed in steps of 4 through v_wmma_f32_16x16x4_f32.
// A layout: lane (m = lane&15, kh = lane>>4): v0 = A[m][k0+2kh], v1 = A[m][k0+2kh+1]
// B layout: v0 = B[k0+2kh][n], v1 = B[k0+2kh+1][n], n = nt*16 + m
// C/D layout: lane, vgpr j -> row (kh*8 + j), col (nt*16 + m)
// ---------------------------------------------------------------------------
__global__ void __launch_bounds__(128)
router_update_kernel(const float* __restrict__ router_feat,
                     const float* __restrict__ Wr,   // (256,128) row-major
                     const float* __restrict__ br,   // (128,)
                     const float* __restrict__ h2,   // (NR,128) in ws
                     float* __restrict__ io,         // (NR,128): h1 in, out
                     int numTiles) {
  int lane = threadIdx.x & 31;
  int wave = threadIdx.x >> 5;
  int tile = (int)blockIdx.x * 4 + wave;
  if (tile >= numTiles) return;

  int m = lane & 15;
  int kh = lane >> 4;  // 0 or 1

  const float* h1row = io + (long)(tile * 16 + m) * HD;
  const float* h2row = h2 + (long)(tile * 16 + m) * HD;

  v8f zero8 = {0.f, 0.f, 0.f, 0.f, 0.f, 0.f, 0.f, 0.f};
  v8f acc[8];
#pragma unroll
  for (int nt = 0; nt < 8; ++nt) acc[nt] = zero8;

  // First half of concat: h1 @ Wr[0:128,:]
  for (int k0 = 0; k0 < HD; k0 += 4) {
    int kk = k0 + kh * 2;
    v2f a;
    a.x = h1row[kk];
    a.y = h1row[kk + 1];
    const float* w = Wr + (long)kk * HD;
#pragma unroll
    for (int nt = 0; nt < 8; ++nt) {
      v2f b;
      b.x = w[nt * 16 + m];
      b.y = w[HD + nt * 16 + m];
      acc[nt] = __builtin_amdgcn_wmma_f32_16x16x4_f32(
          false, a, false, b, (short)0, acc[nt], false, false);
    }
  }
  // Second half of concat: h2 @ Wr[128:256,:]
  for (int k0 = 0; k0 < HD; k0 += 4) {
    int kk = k0 + kh * 2;
    v2f a;
    a.x = h2row[kk];
    a.y = h2row[kk + 1];
    const float* w = Wr + (long)(HD + kk) * HD;
#pragma unroll
    for (int nt = 0; nt < 8; ++nt) {
      v2f b;
      b.x = w[nt * 16 + m];
      b.y = w[HD + nt * 16 + m];
      acc[nt] = __builtin_amdgcn_wmma_f32_16x16x4_f32(
          false, a, false, b, (short)0, acc[nt], false, false);
    }
  }

  // Epilogue: bias + relu + residual; overwrite this tile's rows in place.
#pragma unroll
  for (int nt = 0; nt < 8; ++nt) {
    int col = nt * 16 + m;
    float bias = br[col];
#pragma unroll
    for (int j = 0; j < 8; ++j) {
      int row = tile * 16 + kh * 8 + j;
      float v = acc[nt][j] + bias;
      v = v > 0.f ? v : 0.f;
      long idx = (long)row * HD + col;
      io[idx] = router_feat[idx] + v;
    }
  }
}

// ---------------------------------------------------------------------------
// Packet update: out = feat + relu((s / max(cnt,1)) @ W_p + b_p)
// io holds the transfer sum s on entry (accumulated in d_out), output in place.
// ---------------------------------------------------------------------------
__global__ void __launch_bounds__(128)
packet_update_kernel(const float* __restrict__ packet_feat,
                     const float* __restrict__ Wp,   // (128,128) row-major
                     const float* __restrict__ bp,   // (128,)
                     const float* __restrict__ cnt,  // (NP,)
                     float* __restrict__ io,         // (NP,128): sum in, out
                     int numTiles) {
  int lane = threadIdx.x & 31;
  int wave = threadIdx.x >> 5;
  int tile = (int)blockIdx.x * 4 + wave;
  if (tile >= numTiles) return;

  int m = lane & 15;
  int kh = lane >> 4;

  float c = cnt[tile * 16 + m];
  float inv = 1.0f / fmaxf(c, 1.0f);
  const float* srow = io + (long)(tile * 16 + m) * HD;

  v8f zero8 = {0.f, 0.f, 0.f, 0.f, 0.f, 0.f, 0.f, 0.f};
  v8f acc[8];
#pragma unroll
  for (int nt = 0; nt < 8; ++nt) acc[nt] = zero8;

  for (int k0 = 0; k0 < HD; k0 += 4) {
    int kk = k0 + kh * 2;
    v2f a;
    a.x = srow[kk] * inv;
    a.y = srow[kk + 1] * inv;
    const float* w = Wp + (long)kk * HD;
#pragma unroll
    for (int nt = 0; nt < 8; ++nt) {
      v2f b;
      b.x = w[nt * 16 + m];
      b.y = w[HD + nt * 16 + m];
      acc[nt] = __builtin_amdgcn_wmma_f32_16x16x4_f32(
          false, a, false, b, (short)0, acc[nt], false, false);
    }
  }

#pragma unroll
  for (int nt = 0; nt < 8; ++nt) {
    int col = nt * 16 + m;
    float bias = bp[col];
#pragma unroll
    for (int j = 0; j < 8; ++j) {
      int row = tile * 16 + kh * 8 + j;
      float v = acc[nt][j] + bias;
      v = v > 0.f ? v : 0.f;
      long idx = (long)row * HD + col;
      io[idx] = packet_feat[idx] + v;
    }
  }
}

// ---------------------------------------------------------------------------
extern "C" void kernel_launch(void* const* d_in, const int* in_sizes, int n_in,
                              void* d_out, int out_size, void* d_ws, size_t ws_size,
                              hipStream_t stream) {
  const float* router_feat = (const float*)d_in[0];
  const float* packet_feat = (const float*)d_in[1];
  const float* W_r = (const float*)d_in[2];
  const float* b_r = (const float*)d_in[3];
  const float* W_p = (const float*)d_in[4];
  const float* b_p = (const float*)d_in[5];
  const int* pass_src = (const int*)d_in[6];
  const int* pass_dst = (const int*)d_in[7];
  const int* conn_src = (const int*)d_in[8];
  const int* conn_dst = (const int*)d_in[9];
  const int* tr_src = (const int*)d_in[10];
  const int* tr_dst = (const int*)d_in[11];

  const int NR = in_sizes[0] / HD;   // 100000
  const int NP = in_sizes[1] / HD;   // 200000
  const int E_pass = in_sizes[6];
  const int E_conn = in_sizes[8];
  const int E_tr = in_sizes[10];

  float* out_router = (float*)d_out;                    // h1 accum -> router_out
  float* out_packet = (float*)d_out + (long)NR * HD;    // sum accum -> packet_out
  float* h2 = (float*)d_ws;                             // NR*HD floats
  float* cnt = (float*)d_ws + (long)NR * HD;            // NP floats

  // Zero accumulators (d_out entire + ws regions) every launch: deterministic.
  long n_out4 = ((long)NR * HD + (long)NP * HD) / 4;
  long n_ws4 = ((long)NR * HD + (long)NP) / 4;
  zero_f4_kernel<<<2048, 256, 0, stream>>>((float4*)d_out, n_out4);
  zero_f4_kernel<<<1024, 256, 0, stream>>>((float4*)d_ws, n_ws4);

  // Edge scatters: 8 waves (edges) per 256-thread block.
  int blk;
  blk = (E_pass + 7) / 8;
  scatter_add_kernel<<<blk, 256, 0, stream>>>(router_feat, pass_src, pass_dst,
                                              out_router, nullptr, E_pass);
  blk = (E_conn + 7) / 8;
  scatter_add_kernel<<<blk, 256, 0, stream>>>(packet_feat, conn_src, conn_dst,
                                              h2, nullptr, E_conn);
  blk = (E_tr + 7) / 8;
  scatter_add_kernel<<<blk, 256, 0, stream>>>(router_feat, tr_src, tr_dst,
                                              out_packet, cnt, E_tr);

  // Dense updates via f32 WMMA.
  int rTiles = NR / 16;  // NR divisible by 16
  router_update_kernel<<<(rTiles + 3) / 4, 128, 0, stream>>>(
      router_feat, W_r, b_r, h2, out_router, rTiles);
  int pTiles = NP / 16;
  packet_update_kernel<<<(pTiles + 3) / 4, 128, 0, stream>>>(
      packet_feat, W_p, b_p, cnt, out_packet, pTiles);
}